// POISEVAE_58944131170988
// MI455X (gfx1250) — compile-verified
//
#include <hip/hip_runtime.h>
#include <hip/hip_bf16.h>

typedef __attribute__((ext_vector_type(2))) float v2f;
typedef __attribute__((ext_vector_type(8))) float v8f;

#define D_DIM 8192
#define L_DIM 64
#define B_DIM 1024

static __device__ __forceinline__ v8f wmma_f32(v2f a, v2f b, v8f c) {
    return __builtin_amdgcn_wmma_f32_16x16x4_f32(false, a, false, b, (short)0, c, false, false);
}

// Counter-based RNG: splitmix64 -> Box-Muller pair. Deterministic per counter.
static __device__ __forceinline__ float2 nrm2(unsigned long long x) {
    x += 0x9E3779B97F4A7C15ull;
    x = (x ^ (x >> 30)) * 0xBF58476D1CE4E5B9ull;
    x = (x ^ (x >> 27)) * 0x94D049BB133111EBull;
    x ^= (x >> 31);
    unsigned lo = (unsigned)x, hi = (unsigned)(x >> 32);
    float u1 = (float)(lo >> 8) * 0x1.0p-24f + 0x1.0p-25f;   // (0,1)
    float u2 = (float)(hi >> 8) * 0x1.0p-24f;                // [0,1)
    float r  = sqrtf(-2.0f * __logf(u1));
    float ang = 6.28318530717958647f * u2;
    return make_float2(r * __cosf(ang), r * __sinf(ang));
}

// ---------------------------------------------------------------------------
// Prep: swizzle g11 / g22n(=-exp(g22)) into exact WMMA B-operand layout blobs,
// both orientations, and zero the partial-sum buffer.
// Blob b, chunk c (K=4c..4c+3), tile t (N=16t..16t+15); per 256B block,
// lane L holds pair (K=4c+2*(L>>4)+0, K=...+1) at column N=16t+(L&15).
// blob0: B[k][n]=g11[n][k]   (z2 @ g11^T)
// blob1: B[k][n]=g11[k][n]   (z1 @ g11)
// blob2: B[k][n]=g22n[n][k]  ((z2^2) @ g22n^T)
// blob3: B[k][n]=g22n[k][n]  ((z1^2) @ g22n)
// ---------------------------------------------------------------------------
__global__ __launch_bounds__(256)
void prep_kernel(const float* __restrict__ g11, const float* __restrict__ g22,
                 float* __restrict__ GB, float* __restrict__ part) {
    int e = blockIdx.x * 256 + threadIdx.x;          // 16384 total
    if (e < 256) part[e] = 0.0f;
    int blob = e >> 12, rem = e & 4095;
    int blk  = rem >> 6, pr = rem & 63;
    int lane = pr >> 1, slot = pr & 1;
    int c = blk >> 2, t = blk & 3;
    int k = 4 * c + 2 * (lane >> 4) + slot;
    int n = 16 * t + (lane & 15);
    float v;
    if      (blob == 0) v = g11[n * 64 + k];
    else if (blob == 1) v = g11[k * 64 + n];
    else if (blob == 2) v = -__expf(g22[n * 64 + k]);
    else                v = -__expf(g22[k * 64 + n]);
    GB[e] = v;
}

// ---------------------------------------------------------------------------
// Encoder: one wave per 16-row tile; fused mu + log_var GEMMs (x read once).
// K=8192 -> 2048 chunks of WMMA f32 16x16x4. Outputs mu and var=-exp(lv).
// ---------------------------------------------------------------------------
__global__ __launch_bounds__(32)
void encoder_kernel(const float* __restrict__ x,
                    const float* __restrict__ Wmu, const float* __restrict__ bmu,
                    const float* __restrict__ Wlv, const float* __restrict__ blv,
                    float* __restrict__ outMu, float* __restrict__ outVar) {
    const int lane = threadIdx.x;
    const int m0 = blockIdx.x * 16;
    const int lrow = lane & 15, hl = lane >> 4;
    v8f accMu[4], accLv[4];
#pragma unroll
    for (int t = 0; t < 4; ++t) { accMu[t] = {}; accLv[t] = {}; }
    const float* arow = x + (size_t)(m0 + lrow) * D_DIM + 2 * hl;
    for (int c = 0; c < 2048; ++c) {
        v2f a = *(const v2f*)(arow + 4 * c);
        int k0 = 4 * c + 2 * hl;
#pragma unroll
        for (int t = 0; t < 4; ++t) {
            int n = lrow + 16 * t;
            v2f bm; bm.x = Wmu[k0 * 64 + n]; bm.y = Wmu[(k0 + 1) * 64 + n];
            accMu[t] = wmma_f32(a, bm, accMu[t]);
            v2f bl; bl.x = Wlv[k0 * 64 + n]; bl.y = Wlv[(k0 + 1) * 64 + n];
            accLv[t] = wmma_f32(a, bl, accLv[t]);
        }
    }
#pragma unroll
    for (int t = 0; t < 4; ++t) {
        int n = lrow + 16 * t;
        float bM = bmu[n], bL = blv[n];
#pragma unroll
        for (int r = 0; r < 8; ++r) {
            int M = m0 + r + 8 * hl;
            outMu[M * 64 + n]  = accMu[t][r] + bM;
            outVar[M * 64 + n] = -__expf(accLv[t][r] + bL);
        }
    }
}

// ---------------------------------------------------------------------------
// Gibbs chains: one wave32 per (16-row group, chain). Whole chain runs with
// z-state + mu/var + pre-swizzled G blobs in LDS; no barriers (single wave).
// Each half-step: A-layout ds_load_b64 of z, square in regs, 128 WMMAs
// (eta1 via g11-blob, eta2 via g22-blob), Gaussian resample, C-layout store.
// ---------------------------------------------------------------------------
__global__ __launch_bounds__(32)
void gibbs_kernel(const float* __restrict__ MU1, const float* __restrict__ VAR1,
                  const float* __restrict__ MU2, const float* __restrict__ VAR2,
                  const float* __restrict__ GB, float* __restrict__ Z,
                  const int* __restrict__ pNinit, const int* __restrict__ pNsamp) {
    extern __shared__ float smem[];
    float* sG  = smem;             // 16384 floats: 4 blobs
    float* sZ0 = smem + 16384;     // 1024: z1 (16x64 row-major)
    float* sZ1 = smem + 17408;     // 1024: z2
    float* sMV = smem + 18432;     // 4096: mu1|var1|mu2|var2

    const int lane = threadIdx.x;
    const int mt = blockIdx.x;     // row group 0..63
    const int chain = blockIdx.y;  // 0=prior, 1=posterior
    const int m0 = mt * 16;
    const int lrow = lane & 15, hl = lane >> 4;

    for (int i = lane; i < 16384; i += 32) sG[i] = GB[i];
    if (chain) {
        for (int i = lane; i < 1024; i += 32) {
            int r = i >> 6, cc = i & 63;
            int g = (m0 + r) * 64 + cc;
            sMV[i]        = MU1[g];  sMV[1024 + i] = VAR1[g];
            sMV[2048 + i] = MU2[g];  sMV[3072 + i] = VAR2[g];
        }
    }
    // z0 ~ N(0,1), identical across both chains (reference shares z0).
    for (int i = 2 * lane; i < 1024; i += 64) {
        unsigned eg = (unsigned)((m0 + (i >> 6)) << 6 | (i & 63));
        float2 n1 = nrm2(((unsigned long long)1 << 32) | eg);
        float2 n2 = nrm2(((unsigned long long)2 << 32) | eg);
        sZ0[i] = n1.x; sZ0[i + 1] = n1.y;
        sZ1[i] = n2.x; sZ1[i + 1] = n2.y;
    }

    const int nTot = pNinit[0] + pNsamp[0];
    for (int it = 0; it < nTot; ++it) {
#pragma unroll
        for (int half = 0; half < 2; ++half) {
            const float* zsrc = half ? sZ0 : sZ1;   // update z1 from z2, then z2 from z1
            float*       zdst = half ? sZ1 : sZ0;
            const float* Blin = sG + (half ? 4096 : 0);       // g11-orient blob
            const float* Bsq  = sG + (half ? 12288 : 8192);   // g22n-orient blob
            v2f a[16], a2[16];
#pragma unroll
            for (int c = 0; c < 16; ++c) {
                a[c]  = *(const v2f*)&zsrc[lrow * 64 + 4 * c + 2 * hl];
                a2[c] = a[c] * a[c];
            }
#pragma unroll
            for (int t = 0; t < 4; ++t) {
                v8f acc1 = {}, acc2 = {};
#pragma unroll
                for (int c = 0; c < 16; ++c) {
                    v2f b1 = *(const v2f*)&Blin[(c * 4 + t) * 64 + lane * 2];
                    acc1 = wmma_f32(a[c], b1, acc1);
                    v2f b2 = *(const v2f*)&Bsq[(c * 4 + t) * 64 + lane * 2];
                    acc2 = wmma_f32(a2[c], b2, acc2);
                }
                const float* smu = sMV + half * 2048;
                const float* svr = sMV + half * 2048 + 1024;
#pragma unroll
                for (int r = 0; r < 8; r += 2) {
                    unsigned long long ctr =
                        ((unsigned long long)(0x10000000u + (unsigned)it * 4u +
                                              (unsigned)half * 2u + (unsigned)chain) << 32) |
                        (unsigned)(((m0 + r + 8 * hl) << 6) | (lrow + 16 * t));
                    float2 nn = nrm2(ctr);
#pragma unroll
                    for (int rr = 0; rr < 2; ++rr) {
                        int M = r + rr + 8 * hl;
                        int N = lrow + 16 * t;
                        float mu, var;
                        if (chain) { mu = smu[M * 64 + N]; var = svr[M * 64 + N]; }
                        else       { mu = 0.0f;            var = -0.5f; }
                        float e2 = var + acc2[r + rr];          // < 0
                        float vv = -0.5f / e2;
                        float mm = (mu + acc1[r + rr]) * vv;
                        float nv = rr ? nn.y : nn.x;
                        zdst[M * 64 + N] = mm + sqrtf(vv) * nv;
                    }
                }
            }
        }
    }
    float* z1o = Z + (size_t)(chain * 2 + 0) * 65536;
    float* z2o = Z + (size_t)(chain * 2 + 1) * 65536;
    for (int i = lane; i < 1024; i += 32) {
        int r = i >> 6, cc = i & 63;
        z1o[(m0 + r) * 64 + cc] = sZ0[i];
        z2o[(m0 + r) * 64 + cc] = sZ1[i];
    }
}

// ---------------------------------------------------------------------------
// Fused decoder + sum-squared-error: never materializes x_rec (saves 64MB of
// HBM writes + 64MB reads). One wave per (16-row tile, modality); K=64 WMMA.
// ---------------------------------------------------------------------------
__global__ __launch_bounds__(32)
void decoder_kernel(const float* __restrict__ Z,
                    const float* __restrict__ Wd1, const float* __restrict__ bd1,
                    const float* __restrict__ Wd2, const float* __restrict__ bd2,
                    const float* __restrict__ x1, const float* __restrict__ x2,
                    float* __restrict__ part) {
    const int lane = threadIdx.x;
    const int m0 = blockIdx.x * 16;
    const int grp = blockIdx.y;
    const int lrow = lane & 15, hl = lane >> 4;
    const float* Wd = grp ? Wd2 : Wd1;
    const float* bd = grp ? bd2 : bd1;
    const float* x  = grp ? x2  : x1;
    const float* zg = Z + (size_t)(2 + grp) * 65536;   // posterior chain

    v2f a[16];
#pragma unroll
    for (int c = 0; c < 16; ++c)
        a[c] = *(const v2f*)&zg[(m0 + lrow) * 64 + 4 * c + 2 * hl];

    float sum = 0.0f;
    for (int nt = 0; nt < 512; ++nt) {
        int n = nt * 16 + lrow;
        v8f acc = {};
#pragma unroll
        for (int c = 0; c < 16; ++c) {
            int k0 = 4 * c + 2 * hl;
            v2f b; b.x = Wd[(size_t)k0 * D_DIM + n]; b.y = Wd[(size_t)(k0 + 1) * D_DIM + n];
            acc = wmma_f32(a[c], b, acc);
        }
        float bias = bd[n];
#pragma unroll
        for (int r = 0; r < 8; ++r) {
            float d = acc[r] + bias - x[(size_t)(m0 + r + 8 * hl) * D_DIM + n];
            sum += d * d;
        }
    }
    for (int off = 16; off > 0; off >>= 1) sum += __shfl_down(sum, off, 32);
    if (lane == 0) part[grp * 64 + blockIdx.x] = sum;
}

// ---------------------------------------------------------------------------
// KL estimate: sum mu*(zq-zp) + var*(zq^2-zp^2) over both latent groups.
// ---------------------------------------------------------------------------
__global__ __launch_bounds__(256)
void kl_kernel(const float* __restrict__ MU1, const float* __restrict__ VAR1,
               const float* __restrict__ MU2, const float* __restrict__ VAR2,
               const float* __restrict__ Z, float* __restrict__ part) {
    int tid = blockIdx.x * 256 + threadIdx.x;    // 16384 threads
    float s = 0.0f;
    for (int e = tid; e < 65536; e += 16384) {
        float zq = Z[2 * 65536 + e], zp = Z[0 * 65536 + e];
        s += MU1[e] * (zq - zp) + VAR1[e] * (zq * zq - zp * zp);
        zq = Z[3 * 65536 + e]; zp = Z[1 * 65536 + e];
        s += MU2[e] * (zq - zp) + VAR2[e] * (zq * zq - zp * zp);
    }
    __shared__ float red[256];
    red[threadIdx.x] = s; __syncthreads();
    for (int off = 128; off > 0; off >>= 1) {
        if (threadIdx.x < off) red[threadIdx.x] += red[threadIdx.x + off];
        __syncthreads();
    }
    if (threadIdx.x == 0) part[128 + blockIdx.x] = red[0];
}

__global__ __launch_bounds__(256)
void final_kernel(const float* __restrict__ part, float* __restrict__ out) {
    __shared__ float red[256];
    red[threadIdx.x] = (threadIdx.x < 192) ? part[threadIdx.x] : 0.0f;
    __syncthreads();
    for (int off = 128; off > 0; off >>= 1) {
        if (threadIdx.x < off) red[threadIdx.x] += red[threadIdx.x + off];
        __syncthreads();
    }
    if (threadIdx.x == 0) out[0] = red[0];
}

extern "C" void kernel_launch(void* const* d_in, const int* in_sizes, int n_in,
                              void* d_out, int out_size, void* d_ws, size_t ws_size,
                              hipStream_t stream) {
    const float* x1     = (const float*)d_in[0];
    const float* x2     = (const float*)d_in[1];
    const float* We1_mu = (const float*)d_in[2];
    const float* be1_mu = (const float*)d_in[3];
    const float* We1_lv = (const float*)d_in[4];
    const float* be1_lv = (const float*)d_in[5];
    const float* We2_mu = (const float*)d_in[6];
    const float* be2_mu = (const float*)d_in[7];
    const float* We2_lv = (const float*)d_in[8];
    const float* be2_lv = (const float*)d_in[9];
    const float* Wd1    = (const float*)d_in[10];
    const float* bd1    = (const float*)d_in[11];
    const float* Wd2    = (const float*)d_in[12];
    const float* bd2    = (const float*)d_in[13];
    const float* g11    = (const float*)d_in[14];
    const float* g22    = (const float*)d_in[15];
    const int*   nInit  = (const int*)d_in[16];
    const int*   nSamp  = (const int*)d_in[17];

    float* W    = (float*)d_ws;
    float* MU1  = W;               // 65536
    float* VAR1 = W + 65536;
    float* MU2  = W + 131072;
    float* VAR2 = W + 196608;
    float* Z    = W + 262144;      // 4 x 65536 (z1pri,z2pri,z1pos,z2pos)
    float* GB   = W + 524288;      // 16384 swizzled B-layout blobs
    float* PART = W + 540672;      // 256 partials

    prep_kernel<<<64, 256, 0, stream>>>(g11, g22, GB, PART);
    encoder_kernel<<<64, 32, 0, stream>>>(x1, We1_mu, be1_mu, We1_lv, be1_lv, MU1, VAR1);
    encoder_kernel<<<64, 32, 0, stream>>>(x2, We2_mu, be2_mu, We2_lv, be2_lv, MU2, VAR2);
    gibbs_kernel<<<dim3(64, 2), 32, 22528 * sizeof(float), stream>>>(
        MU1, VAR1, MU2, VAR2, GB, Z, nInit, nSamp);
    decoder_kernel<<<dim3(64, 2), 32, 0, stream>>>(Z, Wd1, bd1, Wd2, bd2, x1, x2, PART);
    kl_kernel<<<64, 256, 0, stream>>>(MU1, VAR1, MU2, VAR2, Z, PART);
    final_kernel<<<1, 256, 0, stream>>>(PART, (float*)d_out);
}